// GridSamplePScan_65687229825295
// MI455X (gfx1250) — compile-verified
//
#include <hip/hip_runtime.h>
#include <cstdint>

// ---------------------------------------------------------------------------
// GridSamplePScan for MI455X (gfx1250, wave32).  B=2, L=16, C=16, H=W=64.
//
// Stage 1: cum = cumsum(flows, axis=1) as a lower-triangular 16x16 GEMM using
//          V_WMMA_F32_16X16X4_F32 (exact: triangular weights are 0/1).
// Stage 2: transpose images to channel-last (B,L,H,W,C) so each bilinear
//          corner's 16 channels are one 64B cacheline -> b128 gathers.
// Stage 3: per (b,l,256-px tile) block: async-stage the 32KB cum tile into
//          LDS (ASYNCcnt path), then per pixel loop k<=l computing bilinear
//          weights once, gathering 4x float4 per corner, 16 channel accs.
// Fallback (tiny workspace): fused kernel, cumsum of async-staged flows tile
//          done in LDS, gathers from original layout.
// ---------------------------------------------------------------------------

typedef __attribute__((ext_vector_type(2))) float v2f;
typedef __attribute__((ext_vector_type(8))) float v8f;

#define HW   4096
#define WW   64
#define LL_  16
#define CC   16

// ---------------- Stage 1: causal prefix sum via WMMA ----------------------
// flows/cum layout: (B, L, 2, H, W) f32, flat = (b*32 + l*2 + d)*4096 + hw.
// Positions p = (b,d,hw) flattened: P = 16384; 1024 waves, 16 positions each.
__global__ __launch_bounds__(256)
void pscan_cumsum_wmma(const float* __restrict__ flows, float* __restrict__ cum) {
    const int lane   = threadIdx.x & 31;
    const int waveId = (blockIdx.x << 3) + (threadIdx.x >> 5);   // 0..1023
    const int n      = lane & 15;       // N column (position within tile)
    const int half   = lane >> 4;       // lane half selects K pair / M block
    const int p0     = waveId << 4;
    const int b      = p0 >> 13;
    const int d      = (p0 >> 12) & 1;
    const int hw     = (p0 & (HW - 1)) + n;
    const int fbase  = (b * 32 + d) * HW + hw;    // + k*8192 to step k
    const int M      = n;                         // A-matrix row carried by lane

    v8f acc = {};
#pragma unroll
    for (int kb = 0; kb < 4; ++kb) {
        const int k0 = kb * 4 + half * 2;
        v2f a, bm;
        a.x = (M >= k0)     ? 1.0f : 0.0f;        // T[M,k] = (M >= k)
        a.y = (M >= k0 + 1) ? 1.0f : 0.0f;
        bm.x = flows[fbase + k0       * (2 * HW)];
        bm.y = flows[fbase + (k0 + 1) * (2 * HW)];
        acc = __builtin_amdgcn_wmma_f32_16x16x4_f32(
            false, a, false, bm, (short)0, acc, false, false);
    }
#pragma unroll
    for (int r = 0; r < 8; ++r) {
        const int l = r + half * 8;
        cum[(b * 32 + l * 2 + d) * HW + hw] = acc[r];
    }
}

// ---------------- Stage 2: images (B,L,C,H,W) -> imgT (B,L,H,W,C) ----------
__global__ __launch_bounds__(256)
void transpose_images(const float* __restrict__ images, float* __restrict__ imgT) {
    const int tid  = threadIdx.x;
    const int blk  = blockIdx.x;        // 512 = (b*16+l)*16 + tile
    const int tile = blk & 15;
    const int bl   = blk >> 4;          // b*16 + l
    const int hw0  = tile << 8;

    const float* src = images + (size_t)bl * (CC * HW) + hw0;   // + c*HW + tid
    float v[CC];
#pragma unroll
    for (int c = 0; c < CC; ++c) v[c] = src[c * HW + tid];      // coalesced per c

    float4* dst = (float4*)(imgT + ((size_t)bl * HW + hw0) * CC);
#pragma unroll
    for (int q = 0; q < 4; ++q) {
        float4 o;
        o.x = v[4 * q + 0]; o.y = v[4 * q + 1];
        o.z = v[4 * q + 2]; o.w = v[4 * q + 3];
        dst[tid * 4 + q] = o;                                   // coalesced b128
    }
}

// ---------------- shared helpers ------------------------------------------
__device__ __forceinline__ float wrap_pm1(float g) {
    // jnp.mod(g+1, 2) - 1, with the reference's guard.
    float t = g + 1.0f;
    t = t - 2.0f * floorf(t * 0.5f);
    float r = t - 1.0f;
    return (r < -1.0f) ? r + 2.0f : r;
}

// Async-stage a (16 l x 2 d x 256 px) f32 tile (32 KB) into LDS.
// gbase points at element (l=0,d=0,px=0) of the tile; rows are HW apart.
__device__ __forceinline__ void stage_tile_async(const float* gbase, float* scum,
                                                 int tid) {
#pragma unroll
    for (int it = 0; it < 8; ++it) {
        const int tr = (it << 8) + tid;     // float4 transfer id 0..2047
        const int e  = tr << 2;
        const int ll = e >> 9;              // 512 floats per l
        const int dd = (e >> 8) & 1;
        const int px = e & 255;
        const float* gsrc = gbase + ((ll * 2 + dd) * HW + px);
        const uint32_t ldsoff = (uint32_t)(uintptr_t)(&scum[e]);
        asm volatile("global_load_async_to_lds_b128 %0, %1, off"
                     :: "v"(ldsoff), "v"(gsrc) : "memory");
    }
    asm volatile("s_wait_asynccnt 0" ::: "memory");
}

// ---------------- Stage 3: causal bilinear gather (channel-last) -----------
__global__ __launch_bounds__(256)
void pscan_sample_t(const float* __restrict__ imgT,
                    const float* __restrict__ cum,
                    float* __restrict__ out) {
    __shared__ __align__(16) float scum[LL_ * 2 * 256];   // [l][d][px] = 32 KB

    const int tid  = threadIdx.x;
    const int blk  = blockIdx.x;       // (b, l, tile)
    const int tile = blk & 15;
    const int l    = (blk >> 4) & 15;
    const int b    = blk >> 8;
    const int hw0  = tile << 8;

    stage_tile_async(cum + (size_t)(b * 32) * HW + hw0, scum, tid);
    __syncthreads();

    const int hw = hw0 + tid;
    const int x  = hw & (WW - 1);
    const int y  = hw >> 6;
    const float bx = ((float)x + 0.5f) * (2.0f / 64.0f) - 1.0f;
    const float by = ((float)y + 0.5f) * (2.0f / 64.0f) - 1.0f;
    const float clx = scum[(l * 2 + 0) * 256 + tid];
    const float cly = scum[(l * 2 + 1) * 256 + tid];

    float acc[CC];
#pragma unroll
    for (int c = 0; c < CC; ++c) acc[c] = 0.0f;

    const float* imgB = imgT + (size_t)(b * LL_) * (HW * CC);

    for (int k = 0; k <= l; ++k) {
        const float gx = wrap_pm1(bx + clx - scum[(k * 2 + 0) * 256 + tid]);
        const float gy = wrap_pm1(by + cly - scum[(k * 2 + 1) * 256 + tid]);
        const float rx = gx * 32.0f + 31.5f;   // (gx+1)*0.5*W - 0.5
        const float ry = gy * 32.0f + 31.5f;
        const float fx0 = floorf(rx);
        const float fy0 = floorf(ry);
        const int x0 = (int)fx0, y0 = (int)fy0;
        const int x1 = x0 + 1,   y1 = y0 + 1;
        const float fx = rx - fx0, fy = ry - fy0;
        const float mx0 = (x0 >= 0 && x0 < WW) ? 1.0f : 0.0f;
        const float mx1 = (x1 >= 0 && x1 < WW) ? 1.0f : 0.0f;
        const float my0 = (y0 >= 0 && y0 < WW) ? 1.0f : 0.0f;
        const float my1 = (y1 >= 0 && y1 < WW) ? 1.0f : 0.0f;
        const float wa = (1.0f - fx) * (1.0f - fy) * mx0 * my0;  // (x0,y0)
        const float wb = (1.0f - fx) * fy          * mx0 * my1;  // (x0,y1)
        const float wc = fx * (1.0f - fy)          * mx1 * my0;  // (x1,y0)
        const float wd = fx * fy                   * mx1 * my1;  // (x1,y1)
        const int xc0 = min(max(x0, 0), WW - 1);
        const int xc1 = min(max(x1, 0), WW - 1);
        const int yc0 = min(max(y0, 0), WW - 1);
        const int yc1 = min(max(y1, 0), WW - 1);
        const float* img = imgB + (size_t)k * (HW * CC);
        const float4* p00 = (const float4*)(img + (size_t)(yc0 * WW + xc0) * CC);
        const float4* p01 = (const float4*)(img + (size_t)(yc1 * WW + xc0) * CC);
        const float4* p10 = (const float4*)(img + (size_t)(yc0 * WW + xc1) * CC);
        const float4* p11 = (const float4*)(img + (size_t)(yc1 * WW + xc1) * CC);
#pragma unroll
        for (int q = 0; q < 4; ++q) {
            const float4 v00 = p00[q], v01 = p01[q], v10 = p10[q], v11 = p11[q];
            acc[4 * q + 0] += wa * v00.x + wb * v01.x + wc * v10.x + wd * v11.x;
            acc[4 * q + 1] += wa * v00.y + wb * v01.y + wc * v10.y + wd * v11.y;
            acc[4 * q + 2] += wa * v00.z + wb * v01.z + wc * v10.z + wd * v11.z;
            acc[4 * q + 3] += wa * v00.w + wb * v01.w + wc * v10.w + wd * v11.w;
        }
    }

    float* outB = out + ((size_t)(b * LL_ + l) * CC) * HW + hw;   // (B,L,C,H,W)
#pragma unroll
    for (int c = 0; c < CC; ++c) outB[(size_t)c * HW] = acc[c];
}

// ---------------- Fallback: fused, zero-workspace --------------------------
__global__ __launch_bounds__(256)
void pscan_sample_fused(const float* __restrict__ images,
                        const float* __restrict__ flows,
                        float* __restrict__ out) {
    __shared__ __align__(16) float scum[LL_ * 2 * 256];

    const int tid  = threadIdx.x;
    const int blk  = blockIdx.x;
    const int tile = blk & 15;
    const int l    = (blk >> 4) & 15;
    const int b    = blk >> 8;
    const int hw0  = tile << 8;

    stage_tile_async(flows + (size_t)(b * 32) * HW + hw0, scum, tid);
    __syncthreads();
    // cumsum over l in LDS: each thread owns pixel `tid` for both components.
#pragma unroll
    for (int dd = 0; dd < 2; ++dd) {
        float run = 0.0f;
#pragma unroll
        for (int ll = 0; ll < LL_; ++ll) {
            run += scum[(ll * 2 + dd) * 256 + tid];
            scum[(ll * 2 + dd) * 256 + tid] = run;
        }
    }
    __syncthreads();

    const int hw = hw0 + tid;
    const int x  = hw & (WW - 1);
    const int y  = hw >> 6;
    const float bx = ((float)x + 0.5f) * (2.0f / 64.0f) - 1.0f;
    const float by = ((float)y + 0.5f) * (2.0f / 64.0f) - 1.0f;
    const float clx = scum[(l * 2 + 0) * 256 + tid];
    const float cly = scum[(l * 2 + 1) * 256 + tid];

    float acc[CC];
#pragma unroll
    for (int c = 0; c < CC; ++c) acc[c] = 0.0f;

    const float* imgB = images + (size_t)b * (LL_ * CC * HW);
    for (int k = 0; k <= l; ++k) {
        const float gx = wrap_pm1(bx + clx - scum[(k * 2 + 0) * 256 + tid]);
        const float gy = wrap_pm1(by + cly - scum[(k * 2 + 1) * 256 + tid]);
        const float rx = gx * 32.0f + 31.5f;
        const float ry = gy * 32.0f + 31.5f;
        const float fx0 = floorf(rx), fy0 = floorf(ry);
        const int x0 = (int)fx0, y0 = (int)fy0;
        const int x1 = x0 + 1,   y1 = y0 + 1;
        const float fx = rx - fx0, fy = ry - fy0;
        const float mx0 = (x0 >= 0 && x0 < WW) ? 1.0f : 0.0f;
        const float mx1 = (x1 >= 0 && x1 < WW) ? 1.0f : 0.0f;
        const float my0 = (y0 >= 0 && y0 < WW) ? 1.0f : 0.0f;
        const float my1 = (y1 >= 0 && y1 < WW) ? 1.0f : 0.0f;
        const float wa = (1.0f - fx) * (1.0f - fy) * mx0 * my0;
        const float wb = (1.0f - fx) * fy          * mx0 * my1;
        const float wc = fx * (1.0f - fy)          * mx1 * my0;
        const float wd = fx * fy                   * mx1 * my1;
        const int xc0 = min(max(x0, 0), WW - 1);
        const int xc1 = min(max(x1, 0), WW - 1);
        const int yc0 = min(max(y0, 0), WW - 1);
        const int yc1 = min(max(y1, 0), WW - 1);
        const int o00 = yc0 * WW + xc0, o01 = yc1 * WW + xc0;
        const int o10 = yc0 * WW + xc1, o11 = yc1 * WW + xc1;
        const float* img = imgB + (size_t)k * (CC * HW);
#pragma unroll
        for (int c = 0; c < CC; ++c) {
            const float* ic = img + c * HW;
            acc[c] += wa * ic[o00] + wb * ic[o01] + wc * ic[o10] + wd * ic[o11];
        }
    }

    float* outB = out + ((size_t)(b * LL_ + l) * CC) * HW + hw;
#pragma unroll
    for (int c = 0; c < CC; ++c) outB[(size_t)c * HW] = acc[c];
}

// ---------------------------------------------------------------------------
extern "C" void kernel_launch(void* const* d_in, const int* in_sizes, int n_in,
                              void* d_out, int out_size, void* d_ws, size_t ws_size,
                              hipStream_t stream) {
    const float* flows  = (const float*)d_in[0];   // (2,16,2,64,64) f32
    const float* images = (const float*)d_in[1];   // (2,16,16,64,64) f32
    float* out = (float*)d_out;                    // (2,16,16,64,64) f32

    const size_t CUM_BYTES  = (size_t)2 * LL_ * 2 * HW * 4;        // 1 MiB
    const size_t IMGT_BYTES = (size_t)2 * LL_ * CC * HW * 4;       // 8 MiB

    if (ws_size >= CUM_BYTES + IMGT_BYTES) {
        float* cum  = (float*)d_ws;
        float* imgT = (float*)d_ws + (CUM_BYTES / 4);
        pscan_cumsum_wmma<<<128, 256, 0, stream>>>(flows, cum);     // 1024 waves
        transpose_images <<<512, 256, 0, stream>>>(images, imgT);
        pscan_sample_t   <<<512, 256, 0, stream>>>(imgT, cum, out);
    } else {
        pscan_sample_fused<<<512, 256, 0, stream>>>(images, flows, out);
    }
}